// GraphSAGE_91087666413883
// MI455X (gfx1250) — compile-verified
//
#include <hip/hip_runtime.h>

typedef __attribute__((ext_vector_type(2))) float v2f;
typedef __attribute__((ext_vector_type(8))) float v8f;

#define NNODES 100000
#define NEDGES 1600000
#define FW 128   // aggregated feature width (all 3 layers)
#define EPW 4    // edges per wave in scatter (1.6M / 4 = 400k waves, exact)

// ---------------- utility kernels ----------------
__global__ void sage_zero_f32v4(float4* __restrict__ p, int n4) {
  int i = blockIdx.x * blockDim.x + threadIdx.x;
  if (i < n4) p[i] = make_float4(0.f, 0.f, 0.f, 0.f);
}

__global__ void sage_deg(const int* __restrict__ dst, float* __restrict__ deg, int ne) {
  int e = blockIdx.x * blockDim.x + threadIdx.x;
  if (e < ne) atomicAdd(&deg[dst[e]], 1.0f);
}

__global__ void sage_invdeg(float* __restrict__ deg, int n) {
  int i = blockIdx.x * blockDim.x + threadIdx.x;
  if (i < n) deg[i] = 1.0f / fmaxf(deg[i], 1.0f);
}

// ---------------- edge scatter: one wave32 per 4 edges, 128-wide rows -------
// Gather src row (coalesced 512B), scatter-add into dst row via f32 atomics.
// Prefetch all 4 source rows before the atomic loop to overlap gather latency.
__global__ void sage_scatter(const float* __restrict__ feat,
                             const int* __restrict__ src,
                             const int* __restrict__ dst,
                             float* __restrict__ neigh) {
  int wave = (blockIdx.x * blockDim.x + threadIdx.x) >> 5;
  int lane = threadIdx.x & 31;
  int e0 = wave * EPW;
  int s[EPW], d[EPW];
#pragma unroll
  for (int i = 0; i < EPW; ++i) {
    s[i] = src[e0 + i];
    d[i] = dst[e0 + i];
  }
#pragma unroll
  for (int i = 0; i < EPW; ++i)
    __builtin_prefetch(feat + (size_t)s[i] * FW + lane * 4, 0, 0);
#pragma unroll
  for (int i = 0; i < EPW; ++i) {
    const float4 v = *(const float4*)(feat + (size_t)s[i] * FW + lane * 4);
    float* o = neigh + (size_t)d[i] * FW + lane * 4;
    atomicAdd(o + 0, v.x);
    atomicAdd(o + 1, v.y);
    atomicAdd(o + 2, v.z);
    atomicAdd(o + 3, v.w);
  }
}

// ---------------- fused SAGE layer GEMM: out = act(in@Ws + (neigh*inv)@Wn + b)
// One wave owns a full 16-row output stripe (16 x NOUT): A fragments are
// loaded once per k-step and reused across NT = NOUT/16 accumulators
// (independent WMMA chains -> good ILP, 8x less global A traffic).
// V_WMMA_F32_16X16X4_F32 fragment layouts (ISA 7.12.2):
//   A 16x4: lanes 0-15 hold M=0..15 with K={0,1} in v0/v1; lanes 16-31 K={2,3}.
//   B 4x16: lanes 0-15 hold N=0..15 with K={0,1} in v0/v1; lanes 16-31 K={2,3}.
//   C/D:    VGPR r = row M=r (lanes 0-15) / M=r+8 (lanes 16-31), N=lane&15.
template <int NOUT, bool RELU>
__global__ void sage_wmma_gemm(const float* __restrict__ in,     // NNODES x 128
                               const float* __restrict__ neigh,  // NNODES x 128
                               const float* __restrict__ inv,    // NNODES
                               const float* __restrict__ Ws,     // 128 x NOUT
                               const float* __restrict__ Wn,     // 128 x NOUT
                               const float* __restrict__ bias,   // NOUT
                               float* __restrict__ out) {        // NNODES x NOUT
  constexpr int NT = NOUT / 16;
  int wid  = (blockIdx.x * blockDim.x + threadIdx.x) >> 5;
  int lane = threadIdx.x & 31;
  if (wid >= NNODES / 16) return;       // wave-uniform: EXEC stays all-ones
  int m0 = wid * 16;

  int half = lane >> 4;                 // lane half selects K sub-pair
  int idx  = lane & 15;                 // M index (A rows) / N index (B cols)
  int kh   = half << 1;                 // K sub-offset {0,2}

  const float* inRow = in    + (size_t)(m0 + idx) * FW;
  const float* ngRow = neigh + (size_t)(m0 + idx) * FW;
  float scale = inv[m0 + idx];
  const float* wsC = Ws + idx;
  const float* wnC = Wn + idx;

  v8f c[NT];
#pragma unroll
  for (int t = 0; t < NT; ++t) c[t] = (v8f){};

#pragma unroll 2
  for (int k = 0; k < FW; k += 4) {
    v2f as, an;
    as.x = inRow[k + kh];
    as.y = inRow[k + kh + 1];
    an.x = ngRow[k + kh] * scale;
    an.y = ngRow[k + kh + 1] * scale;
    const float* wsK = wsC + (size_t)(k + kh) * NOUT;
    const float* wnK = wnC + (size_t)(k + kh) * NOUT;
#pragma unroll
    for (int t = 0; t < NT; ++t) {
      v2f bs, bn;
      bs.x = wsK[t * 16];
      bs.y = wsK[NOUT + t * 16];
      bn.x = wnK[t * 16];
      bn.y = wnK[NOUT + t * 16];
      c[t] = __builtin_amdgcn_wmma_f32_16x16x4_f32(false, as, false, bs,
                                                   (short)0, c[t], false, false);
      c[t] = __builtin_amdgcn_wmma_f32_16x16x4_f32(false, an, false, bn,
                                                   (short)0, c[t], false, false);
    }
  }

  int mbase = m0 + (half << 3);
#pragma unroll
  for (int t = 0; t < NT; ++t) {
    float bv = bias[t * 16 + idx];
#pragma unroll
    for (int r = 0; r < 8; ++r) {
      float v = c[t][r] + bv;
      if (RELU) v = fmaxf(v, 0.0f);
      out[(size_t)(mbase + r) * NOUT + t * 16 + idx] = v;
    }
  }
}

// ---------------- launcher ----------------
extern "C" void kernel_launch(void* const* d_in, const int* in_sizes, int n_in,
                              void* d_out, int out_size, void* d_ws, size_t ws_size,
                              hipStream_t stream) {
  const float* features = (const float*)d_in[0];
  const int*   src      = (const int*)d_in[1];
  const int*   dst      = (const int*)d_in[2];
  const float* Ws0 = (const float*)d_in[3];
  const float* Wn0 = (const float*)d_in[4];
  const float* b0  = (const float*)d_in[5];
  const float* Ws1 = (const float*)d_in[6];
  const float* Wn1 = (const float*)d_in[7];
  const float* b1  = (const float*)d_in[8];
  const float* Ws2 = (const float*)d_in[9];
  const float* Wn2 = (const float*)d_in[10];
  const float* b2  = (const float*)d_in[11];
  float* out = (float*)d_out;

  float* ws    = (float*)d_ws;
  float* deg   = ws;                                  // NNODES (deg -> inv_deg)
  float* neigh = deg + NNODES;                        // NNODES * 128
  float* h0    = neigh + (size_t)NNODES * FW;         // NNODES * 128
  float* h1    = h0 + (size_t)NNODES * FW;            // NNODES * 128

  const int TB = 256;
  const int nNeigh4 = NNODES * FW / 4;                // 3.2M float4, exact
  dim3 blkZeroDeg((NNODES / 4 + TB - 1) / TB);
  dim3 blkZeroNeigh(nNeigh4 / TB);                    // 12500, exact
  dim3 blkEdge((NEDGES + TB - 1) / TB);
  dim3 blkScatter(NEDGES / EPW * 32 / TB);            // 50000, exact
  dim3 blkInv((NNODES + TB - 1) / TB);
  dim3 blkGemm(((NNODES / 16) * 32 + TB - 1) / TB);   // 6250 waves -> 782 blocks

  // degrees -> inverse degrees (shared by all layers)
  sage_zero_f32v4<<<blkZeroDeg, TB, 0, stream>>>((float4*)deg, NNODES / 4);
  sage_deg<<<blkEdge, TB, 0, stream>>>(dst, deg, NEDGES);
  sage_invdeg<<<blkInv, TB, 0, stream>>>(deg, NNODES);

  // layer 0: features -> h0 (relu)
  sage_zero_f32v4<<<blkZeroNeigh, TB, 0, stream>>>((float4*)neigh, nNeigh4);
  sage_scatter<<<blkScatter, TB, 0, stream>>>(features, src, dst, neigh);
  sage_wmma_gemm<128, true><<<blkGemm, TB, 0, stream>>>(
      features, neigh, deg, Ws0, Wn0, b0, h0);

  // layer 1: h0 -> h1 (relu)
  sage_zero_f32v4<<<blkZeroNeigh, TB, 0, stream>>>((float4*)neigh, nNeigh4);
  sage_scatter<<<blkScatter, TB, 0, stream>>>(h0, src, dst, neigh);
  sage_wmma_gemm<128, true><<<blkGemm, TB, 0, stream>>>(
      h0, neigh, deg, Ws1, Wn1, b1, h1);

  // layer 2: h1 -> out (no relu, width 64)
  sage_zero_f32v4<<<blkZeroNeigh, TB, 0, stream>>>((float4*)neigh, nNeigh4);
  sage_scatter<<<blkScatter, TB, 0, stream>>>(h1, src, dst, neigh);
  sage_wmma_gemm<64, false><<<blkGemm, TB, 0, stream>>>(
      h1, neigh, deg, Ws2, Wn2, b2, out);
}